// ModelNew_36215164240213
// MI455X (gfx1250) — compile-verified
//
#include <hip/hip_runtime.h>
#include <hip/hip_bf16.h>

typedef float v2f __attribute__((ext_vector_type(2)));
typedef float v8f __attribute__((ext_vector_type(8)));

// Problem constants (from reference): b=8, s=4096, h=16, p=64, n=64
#define SB 8
#define SS 4096
#define SH 16
#define SP 64
#define SN 64
#define BH (SB * SH)            // 128 (b,h) pairs
#define KSLICES 8               // K-dim split for occupancy
#define TSLICE (SS / KSLICES)   // 512 timesteps per slice
#define KB 32                   // timesteps staged in LDS per K-block
#define NKB (TSLICE / KB)       // 16 K-blocks per slice
#define LDST 80                 // LDS row stride in floats (conflict-free 2-row frag reads)

// ---------------- Phase 1: per-(b,h) cumsum of A, write suffix-decay d_t ----------------
__global__ void mamba_scan_kernel(const float* __restrict__ A, float* __restrict__ d) {
    __shared__ float sums[256];
    __shared__ float excl[256];
    const int bh = blockIdx.x;          // 0..127
    const int b = bh >> 4, h = bh & 15;
    const int tid = threadIdx.x;        // 256 threads, 16 timesteps each
    const float* Ap = A + ((size_t)b * SS) * SH + h;
    const int tbase = tid * 16;

    float la[16];
    float run = 0.f;
    for (int j = 0; j < 16; ++j) {
        run += Ap[(size_t)(tbase + j) * SH];
        la[j] = run;                    // inclusive local cumsum
    }
    sums[tid] = run;
    __syncthreads();
    if (tid == 0) {
        float acc = 0.f;
        for (int i = 0; i < 256; ++i) { excl[i] = acc; acc += sums[i]; }
    }
    __syncthreads();
    const float total = excl[255] + sums[255];
    const float base  = excl[tid];
    float* dp = d + (size_t)bh * SS;
    for (int j = 0; j < 16; ++j) {
        // d_t = exp( sum of A over (t, end] )  <= 1, underflows to 0 harmlessly
        dp[tbase + j] = expf(total - (base + la[j]));
    }
}

// Per-lane async copy: 16 bytes from global address sbase + goff to the LDS address in lds_off (ASYNCcnt)
__device__ __forceinline__ void async_b128(unsigned lds_off, unsigned goff, const float* sbase) {
    asm volatile("global_load_async_to_lds_b128 %0, %1, %2"
                 :: "v"(lds_off), "v"(goff), "s"(sbase) : "memory");
}
__device__ __forceinline__ void async_b64(unsigned lds_off, unsigned goff, const float* sbase) {
    asm volatile("global_load_async_to_lds_b64 %0, %1, %2"
                 :: "v"(lds_off), "v"(goff), "s"(sbase) : "memory");
}

// ---------------- Phase 2: per-(b,h,kslice) 64x64 WMMA GEMM:  X^T diag(d) B ----------------
__global__ void mamba_gemm_kernel(const float* __restrict__ X,
                                  const float* __restrict__ Bm,
                                  const float* __restrict__ d,
                                  float* __restrict__ part) {
    __shared__ __align__(16) float Xs[2][KB][LDST];   // double-buffered
    __shared__ __align__(16) float Bs[2][KB][LDST];
    __shared__ __align__(16) float dsh[TSLICE];       // decay weights for this slice

    const int blk = blockIdx.x;         // 0..1023
    const int bh = blk >> 3;            // / KSLICES
    const int ks = blk & (KSLICES - 1);
    const int b = bh >> 4, h = bh & 15;
    const int tid  = threadIdx.x;       // 256 = 8 waves
    const int lane = tid & 31;
    const int wave = tid >> 5;
    const int pi = wave & 3;            // output row-tile (p)
    const int nb = (wave >> 2) * 32;    // this wave covers n-tiles nb and nb+16
    const int t0 = ks * TSLICE;

    // Uniform SGPR bases; per-lane VGPR byte offsets (GVS addressing)
    const float* xbase = X  + (size_t)b * SS * SH * SP + (size_t)h * SP;
    const float* bbase = Bm + (size_t)b * SS * SH * SP + (size_t)h * SP;
    const float* dbase = d + (size_t)bh * SS + t0;

    // Low 32 bits of a generic LDS pointer == wave-relative LDS byte offset
    const unsigned xs_lds = (unsigned)(unsigned long long)&Xs[0][0][0];
    const unsigned bs_lds = (unsigned)(unsigned long long)&Bs[0][0][0];
    const unsigned ds_lds = (unsigned)(unsigned long long)&dsh[0];
    const unsigned bufstride = (unsigned)(KB * LDST * 4);

    // This thread's two (row t, col c4) copy slots per 32x64 tile
    const int e0 = tid,        t_0 = e0 >> 4, c4_0 = (e0 & 15) << 2;
    const int e1 = tid + 256,  t_1 = e1 >> 4, c4_1 = (e1 & 15) << 2;

    // Prologue: stage decay weights (2KB) and K-block 0
    async_b64(ds_lds + (unsigned)tid * 8u, (unsigned)tid * 8u, dbase);
    {
        const int tg0 = t0 + t_0, tg1 = t0 + t_1;
        async_b128(xs_lds + (unsigned)((t_0 * LDST + c4_0) * 4), (unsigned)((tg0 * SH * SP + c4_0) * 4), xbase);
        async_b128(bs_lds + (unsigned)((t_0 * LDST + c4_0) * 4), (unsigned)((tg0 * SH * SP + c4_0) * 4), bbase);
        async_b128(xs_lds + (unsigned)((t_1 * LDST + c4_1) * 4), (unsigned)((tg1 * SH * SP + c4_1) * 4), xbase);
        async_b128(bs_lds + (unsigned)((t_1 * LDST + c4_1) * 4), (unsigned)((tg1 * SH * SP + c4_1) * 4), bbase);
    }

    const int m   = lane & 15;
    const int khi = (lane >> 4) * 2;    // A/B frag K offset: lanes 0-15 -> K{0,1}, 16-31 -> K{2,3}

    v8f c0 = {}; v8f c1 = {};

    for (int kb = 0; kb < NKB; ++kb) {
        if (kb + 1 < NKB) {
            // Issue async copy of next K-block into the other buffer
            const unsigned boff = ((kb + 1) & 1) ? bufstride : 0u;
            const int tg0 = t0 + (kb + 1) * KB + t_0;
            const int tg1 = t0 + (kb + 1) * KB + t_1;
            async_b128(xs_lds + boff + (unsigned)((t_0 * LDST + c4_0) * 4), (unsigned)((tg0 * SH * SP + c4_0) * 4), xbase);
            async_b128(bs_lds + boff + (unsigned)((t_0 * LDST + c4_0) * 4), (unsigned)((tg0 * SH * SP + c4_0) * 4), bbase);
            async_b128(xs_lds + boff + (unsigned)((t_1 * LDST + c4_1) * 4), (unsigned)((tg1 * SH * SP + c4_1) * 4), xbase);
            async_b128(bs_lds + boff + (unsigned)((t_1 * LDST + c4_1) * 4), (unsigned)((tg1 * SH * SP + c4_1) * 4), bbase);
            // In-order completion: <=4 outstanding means everything older (incl. block kb) landed
            asm volatile("s_wait_asynccnt 0x4" ::: "memory");
        } else {
            asm volatile("s_wait_asynccnt 0x0" ::: "memory");
        }
        __syncthreads();                 // all waves' async data for block kb visible

        const int buf = kb & 1;
        const int kbt = kb * KB;         // slice-local time base for decay weights
        for (int k = 0; k < KB; k += 4) {
            const float da0 = dsh[kbt + k + khi];
            const float da1 = dsh[kbt + k + khi + 1];
            v2f a, bb0, bb1;
            // A-frag (X^T tile, decay-scaled): lane -> M = pi*16 + m, VGPR r -> K = k + khi + r
            a.x = Xs[buf][k + khi][pi * 16 + m] * da0;
            a.y = Xs[buf][k + khi + 1][pi * 16 + m] * da1;
            // B-frags: lane -> N = nb(+16) + m, VGPR r -> K = k + khi + r
            bb0.x = Bs[buf][k + khi][nb + m];
            bb0.y = Bs[buf][k + khi + 1][nb + m];
            bb1.x = Bs[buf][k + khi][nb + 16 + m];
            bb1.y = Bs[buf][k + khi + 1][nb + 16 + m];
            c0 = __builtin_amdgcn_wmma_f32_16x16x4_f32(false, a, false, bb0, (short)0, c0, false, false);
            c1 = __builtin_amdgcn_wmma_f32_16x16x4_f32(false, a, false, bb1, (short)0, c1, false, false);
        }
        __syncthreads();                 // block kb fully consumed; its buffer may be overwritten
    }

    // write partial 16x16 tiles: C/D layout -> VGPR r, lane L: M = r + 8*(L/16), N = L%16
    float* pout = part + (size_t)blk * (SP * SN);
    const int mh = 8 * (lane >> 4);
    for (int r = 0; r < 8; ++r) {
        const int M = pi * 16 + r + mh;
        pout[M * SN + nb + m]      = c0[r];
        pout[M * SN + nb + 16 + m] = c1[r];
    }
}

// ---------------- Phase 3: reduce K-slice partials ----------------
__global__ void mamba_reduce_kernel(const float* __restrict__ part, float* __restrict__ out) {
    const int i = blockIdx.x * blockDim.x + threadIdx.x;   // 0..524287
    const int bh = i >> 12;          // / 4096
    const int pn = i & 4095;
    float acc = 0.f;
    for (int ks = 0; ks < KSLICES; ++ks)
        acc += part[((size_t)(bh * KSLICES + ks)) * (SP * SN) + pn];
    out[i] = acc;
}

extern "C" void kernel_launch(void* const* d_in, const int* in_sizes, int n_in,
                              void* d_out, int out_size, void* d_ws, size_t ws_size,
                              hipStream_t stream) {
    const float* X  = (const float*)d_in[0];   // (8,4096,16,64)
    const float* A  = (const float*)d_in[1];   // (8,4096,16)
    const float* Bm = (const float*)d_in[2];   // (8,4096,16,64)
    // d_in[3] = C is dead code in the reference output.
    float* out = (float*)d_out;                // (8,16,64,64) = 524288 floats

    float* ws_d    = (float*)d_ws;             // BH*SS floats   = 2 MB
    float* ws_part = ws_d + (size_t)BH * SS;   // BH*KSLICES*4096 floats = 16 MB

    mamba_scan_kernel<<<BH, 256, 0, stream>>>(A, ws_d);
    mamba_gemm_kernel<<<BH * KSLICES, 256, 0, stream>>>(X, Bm, ws_d, ws_part);
    mamba_reduce_kernel<<<(SB * SH * SP * SN) / 256, 256, 0, stream>>>(ws_part, out);
    (void)in_sizes; (void)n_in; (void)out_size; (void)ws_size;
}